// RollingFeatureExtractor_13692355739747
// MI455X (gfx1250) — compile-verified
//
#include <hip/hip_runtime.h>
#include <stdint.h>

// Problem constants (from reference setup_inputs)
#define Bsz 32
#define Ssz 8192
#define Fsz 64
#define Wsz 64

#define TPB  256          // 64 features x 4 output blocks
#define ROWS 320          // 4*64 output rows + 64 halo rows staged in LDS

typedef unsigned int u32x4 __attribute__((ext_vector_type(4)));
typedef int          i32x4 __attribute__((ext_vector_type(4)));
typedef int          i32x8 __attribute__((ext_vector_type(8)));

// out layout per (b,s): [ x(64) | mean(64) | max(64) | min(64) | std(64) ]
__global__ __launch_bounds__(TPB)
void rolling_stats_kernel(const float* __restrict__ x, float* __restrict__ out) {
    extern __shared__ float smem[];
    float* tile   = smem;                         // ROWS * 64 floats   (80 KB)
    float* sufmax = smem + ROWS * Fsz;            // 64 * TPB floats    (64 KB)
    float* sufmin = sufmax + 64 * TPB;            // 64 * TPB floats    (64 KB)

    const int tid      = threadIdx.x;
    const int f        = tid & 63;                // feature
    const int blk      = tid >> 6;                // output block within tile: 0..3
    const int b        = blockIdx.x >> 5;         // batch
    const int tileIdx  = blockIdx.x & 31;         // 32 tiles of 256 positions per batch
    const bool firstTile = (tileIdx == 0);

    const int seqBase  = tileIdx * 256;                       // first output position of WG
    const int loadRow0 = firstTile ? 0   : (seqBase - 64);    // first global seq row staged
    const int dstRow   = firstTile ? 64  : 0;                 // LDS tile row it lands on
    const int nRows    = firstTile ? 256 : 320;
    const size_t gElem0 = ((size_t)b * Ssz + (size_t)loadRow0) * Fsz;

    // ---------------- Stage tile into LDS via Tensor Data Mover ----------------
#if __has_builtin(__builtin_amdgcn_tensor_load_to_lds) && __has_builtin(__builtin_amdgcn_s_wait_tensorcnt)
    if (tid < 32) {   // one wave issues the TDM descriptor for the whole WG
        const uint64_t ga = (uint64_t)(uintptr_t)(x + gElem0);
        const uint32_t la = (uint32_t)(uintptr_t)(tile + (size_t)dstRow * Fsz); // LDS byte offset

        u32x4 g0;
        g0[0] = 1u;                                            // count=1, user mode, no gather
        g0[1] = la;                                            // lds_addr [63:32]
        g0[2] = (uint32_t)ga;                                  // global_addr [95:64]
        g0[3] = (uint32_t)((ga >> 32) & 0x01FFFFFFull)         // global_addr [120:96]
              | (2u << 30);                                    // type=2 ("image")

        i32x8 g1;
        g1[0] = (int)(2u << 16);                               // workgroup_mask=0, data_size=4B
        g1[1] = (int)(64u << 16);                              // tensor_dim0 lo16 = 64 (F)
        g1[2] = (int)(((uint32_t)nRows & 0xFFFFu) << 16);      // dim0 hi16=0 | tensor_dim1 lo16
        g1[3] = (int)(64u << 16);                              // tensor_dim1 hi16=0 | tile_dim0=64
        g1[4] = (int)((uint32_t)nRows & 0xFFFFu);              // tile_dim1=nRows, tile_dim2=0
        g1[5] = 64;                                            // tensor_dim0_stride lo32 = 64 elems
        g1[6] = 0;                                             // stride0 hi16 | stride1 lo16
        g1[7] = 0;

        i32x4 gz = {0, 0, 0, 0};
#if __has_include(<hip/amd_detail/amd_gfx1250_TDM.h>)
        i32x8 gz8 = {0, 0, 0, 0, 0, 0, 0, 0};                  // clang-23 therock 6-arg form
        __builtin_amdgcn_tensor_load_to_lds(g0, g1, gz, gz, gz8, 0);
#else
        __builtin_amdgcn_tensor_load_to_lds(g0, g1, gz, gz, 0); // ROCm 7.2 5-arg form
#endif
        __builtin_amdgcn_s_wait_tensorcnt(0);
    }
#else
    {   // Fallback: cooperative float4 staging
        const float4* src = (const float4*)(x + gElem0);
        float4* dst = (float4*)(tile + (size_t)dstRow * Fsz);
        const int n4 = nRows * (Fsz / 4);
        for (int i = tid; i < n4; i += TPB) dst[i] = src[i];
    }
#endif
    __syncthreads();

    // ---------------- Per-thread sliding-window stats ----------------
    const bool jb0      = firstTile && (blk == 0);  // first block of the batch (replicate pad)
    const int  prowBase = blk * 64;                 // previous-block rows in tile
    const int  crowBase = prowBase + 64;            // current-block rows in tile

    const float cur0 = tile[crowBase * Fsz + f];    // x[b, s0, f] (== x[b,0,f] when jb0)

    // Phase 1: backward suffix scan over the previous block (replicate value if jb0)
    float smax = -__builtin_inff();
    float smin =  __builtin_inff();
    float psum = 0.0f, psumsq = 0.0f, prev0 = 0.0f;
#pragma unroll 8
    for (int i = 63; i >= 0; --i) {
        const float v = jb0 ? cur0 : tile[(prowBase + i) * Fsz + f];
        smax = fmaxf(smax, v);
        smin = fminf(smin, v);
        psum += v;
        psumsq = fmaf(v, v, psumsq);
        sufmax[i * TPB + tid] = smax;               // suffix over [i..63]
        sufmin[i * TPB + tid] = smin;
        prev0 = v;                                  // ends as previous-block row 0
    }
    // sufmax/sufmin entries are private per thread -> no barrier needed (DS in-order per wave)

    // Phase 2: forward pass, O(1) sliding sums + prefix max/min combined with stored suffix
    float wsum   = psum   - prev0 + cur0;           // window for r=0: prev rows 1..63 + cur row 0
    float wsumsq = psumsq - prev0 * prev0 + cur0 * cur0;
    float pmax = cur0, pmin = cur0;

    const uint32_t obase = (uint32_t)(b * Ssz + seqBase + blk * 64) * 320u + (uint32_t)f;
    float* o = out + obase;

#pragma unroll 4
    for (int r = 0; r < 64; ++r) {
        const float xv = tile[(crowBase + r) * Fsz + f];
        if (r > 0) {
            const float lv = jb0 ? cur0 : tile[(prowBase + r) * Fsz + f]; // element leaving window
            wsum   += xv - lv;
            wsumsq += fmaf(xv, xv, -lv * lv);
            pmax = fmaxf(pmax, xv);
            pmin = fminf(pmin, xv);
        }
        float wmax, wmin;
        if (r < 63) {
            wmax = fmaxf(pmax, sufmax[(r + 1) * TPB + tid]);
            wmin = fminf(pmin, sufmin[(r + 1) * TPB + tid]);
        } else {
            wmax = pmax;
            wmin = pmin;
        }
        const float mean = wsum * (1.0f / 64.0f);
        const float var  = (wsumsq - 64.0f * mean * mean) * (1.0f / 63.0f);
        const float sd   = __builtin_sqrtf(fmaxf(var, 0.0f));

        // Write-once output (320 MiB total) -> non-temporal stores so the stream
        // doesn't thrash the 192 MB L2 that adjacent WGs rely on for halo re-reads.
        __builtin_nontemporal_store(xv,   &o[0]);    // pass-through x
        __builtin_nontemporal_store(mean, &o[64]);
        __builtin_nontemporal_store(wmax, &o[128]);
        __builtin_nontemporal_store(wmin, &o[192]);
        __builtin_nontemporal_store(sd,   &o[256]);
        o += 320;
    }
}

extern "C" void kernel_launch(void* const* d_in, const int* in_sizes, int n_in,
                              void* d_out, int out_size, void* d_ws, size_t ws_size,
                              hipStream_t stream) {
    (void)in_sizes; (void)n_in; (void)out_size; (void)d_ws; (void)ws_size;
    const float* x  = (const float*)d_in[0];   // (32, 8192, 64) f32; d_in[1] = window_size (==64)
    float* out      = (float*)d_out;           // (32, 8192, 320) f32

    const size_t ldsBytes = (size_t)(ROWS * Fsz + 2 * 64 * TPB) * sizeof(float); // 208 KB
    // Unconditional (idempotent) host-side attribute set: no static guards, capture-safe.
    hipFuncSetAttribute((const void*)rolling_stats_kernel,
                        hipFuncAttributeMaxDynamicSharedMemorySize, (int)ldsBytes);
    dim3 grid(Bsz * (Ssz / 256));   // 1024 workgroups
    rolling_stats_kernel<<<grid, TPB, ldsBytes, stream>>>(x, out);
}